// APPNP_68289980006843
// MI455X (gfx1250) — compile-verified
//
#include <hip/hip_runtime.h>
#include <cstdint>
#include <cstddef>

// ---------------------------------------------------------------------------
// Temporal LightGCN-style propagation, MI455X (gfx1250).
//
// out layout (== d_out): float out[n_nodes][4][128]   (node-major, matches the
// reference's concat(users, items) of embs stacked on axis=1).
// Slice k holds the UNSCALED k-hop aggregate during propagation; a final
// kernel applies t*(1-t)^k in place. Both the gather slice and the scatter
// slice (76.8 MB each) fit in the 192 MB L2 together; edge streams are loaded
// non-temporally so they don't evict the randomly-reused embedding rows.
//
// NOTE: atomic_probe_kernel is FIRST in this TU purely so the disasm snippet
// shows how unsafeAtomicAdd lowers; spmm_scatter_kernel is second.
// ---------------------------------------------------------------------------

#define DIM 128
#define SLICES 4
#define NODE_STRIDE (SLICES * DIM)   // 512 floats per node in the output tensor
#define CHUNK 256                    // edges staged into LDS per block per stage

// gfx1250 async global->LDS (ASYNCcnt) path, guarded so compilation is safe on
// toolchains where the builtin is absent.
#if defined(__has_builtin)
#  if __has_builtin(__builtin_amdgcn_global_load_async_to_lds_b32) && \
      __has_builtin(__builtin_amdgcn_s_wait_asynccnt)
#    define HAVE_ASYNC_LDS 1
#  endif
#endif
#ifndef HAVE_ASYNC_LDS
#  define HAVE_ASYNC_LDS 0
#endif

#if HAVE_ASYNC_LDS
__device__ __forceinline__ void async_copy_b32_nt(const void* gsrc, void* ldst) {
  // (global int* src, local int* dst, imm offset, imm cpol)
  // cpol = 1 -> TH_NT: edge streams are touched once per hop; keep them from
  // thrashing the L2-resident embedding slices.
  __builtin_amdgcn_global_load_async_to_lds_b32(
      (__attribute__((address_space(1))) int*)(const_cast<void*>(gsrc)),
      (__attribute__((address_space(3))) int*)ldst,
      /*offset=*/0, /*cpol=*/1);
}
#endif

__device__ __forceinline__ void atomic_add_f32(float* p, float v) {
  // Should lower to a single global_atomic_add_f32 (no CAS loop).
  unsafeAtomicAdd(p, v);
}

// ---------------------------------------------------------------------------
// Disasm probe: shows the exact lowering of the f32 atomic add at the top of
// the assembly listing. Never launched.
// ---------------------------------------------------------------------------
__global__ void atomic_probe_kernel(float* p, float v) {
  atomic_add_f32(p + threadIdx.x, v);
}

// ---------------------------------------------------------------------------
// SpMM scatter: one wave32 owns 32 edges; 32 lanes move one 128-float row as
// float4s (coalesced 512B) and accumulate with hardware f32 atomics.
// Edge triples are double-buffer staged into LDS via async loads (ASYNCcnt).
// ---------------------------------------------------------------------------
__global__ __launch_bounds__(256) void spmm_scatter_kernel(
    const int* __restrict__ rows, const int* __restrict__ cols,
    const float* __restrict__ vals,
    const float* __restrict__ src,   // agg slice base (node stride NODE_STRIDE)
    float* __restrict__ dst,         // accumulation slice base
    int n_edges, int n_chunks) {
  __shared__ int   s_rows[2][CHUNK];
  __shared__ int   s_cols[2][CHUNK];
  __shared__ float s_vals[2][CHUNK];

  const int tid  = threadIdx.x;
  const int wave = tid >> 5;
  const int lane = tid & 31;

  int chunk = blockIdx.x;
  if (chunk >= n_chunks) return;

  auto stage = [&](int buf, int base) {
    int e = base + tid;
    if (e < n_edges) {
#if HAVE_ASYNC_LDS
      async_copy_b32_nt(rows + e, &s_rows[buf][tid]);
      async_copy_b32_nt(cols + e, &s_cols[buf][tid]);
      async_copy_b32_nt(vals + e, &s_vals[buf][tid]);
#else
      s_rows[buf][tid] = rows[e];
      s_cols[buf][tid] = cols[e];
      s_vals[buf][tid] = vals[e];
#endif
    }
  };

  int cur = 0;
  stage(0, chunk * CHUNK);

  for (;;) {
    int  next     = chunk + gridDim.x;       // grid-stride over chunks
    bool has_next = (next < n_chunks);
    if (has_next) stage(cur ^ 1, next * CHUNK);  // prefetch next batch

#if HAVE_ASYNC_LDS
    // Async loads complete in order: waiting <=3 (the 3 just-issued next-batch
    // loads) guarantees the current batch is resident in LDS.
    if (has_next) __builtin_amdgcn_s_wait_asynccnt(3);
    else          __builtin_amdgcn_s_wait_asynccnt(0);
#endif
    __syncthreads();

    {
      int ebase = chunk * CHUNK + wave * 32;
      int cnt   = n_edges - ebase;
      cnt = cnt > 32 ? 32 : cnt;
      if (cnt > 0) {
        // Warm L2 for the 32 random gather rows of this wave's batch.
        if (lane < cnt) {
          int c = s_cols[cur][wave * 32 + lane];
          __builtin_prefetch(src + (size_t)c * NODE_STRIDE, 0, 0);
        }
        for (int j = 0; j < cnt; ++j) {
          int   idx = wave * 32 + j;
          int   c   = s_cols[cur][idx];
          int   r   = s_rows[cur][idx];
          float v   = s_vals[cur][idx];
          float4 x  = ((const float4*)(src + (size_t)c * NODE_STRIDE))[lane];
          float* d  = dst + (size_t)r * NODE_STRIDE + (lane << 2);
          atomic_add_f32(d + 0, v * x.x);
          atomic_add_f32(d + 1, v * x.y);
          atomic_add_f32(d + 2, v * x.z);
          atomic_add_f32(d + 3, v * x.w);
        }
      }
    }

    if (!has_next) break;
    __syncthreads();   // everyone done reading `cur` before it is restaged
    cur ^= 1;
    chunk = next;
  }
}

// ---------------------------------------------------------------------------
// init: slice 0 = raw concat(user_embed, item_embed); slices 1..3 = 0
// ---------------------------------------------------------------------------
__global__ __launch_bounds__(256) void init_out_kernel(
    const float4* __restrict__ user_e, const float4* __restrict__ item_e,
    float4* __restrict__ out, int n_users, long long total_f4) {
  long long g = (long long)blockIdx.x * blockDim.x + threadIdx.x;
  if (g >= total_f4) return;
  int node   = (int)(g >> 7);   // 128 float4 per node (4 slices * 32)
  int within = (int)(g & 127);
  int slice  = within >> 5;     // 32 float4 per (node, hop) row
  int d4     = within & 31;
  float4 v = make_float4(0.f, 0.f, 0.f, 0.f);
  if (slice == 0) {
    v = (node < n_users) ? user_e[(long long)node * 32 + d4]
                         : item_e[(long long)(node - n_users) * 32 + d4];
  }
  out[g] = v;
}

// ---------------------------------------------------------------------------
// final: out[n][k][:] *= t_n * (1 - t_n)^k   (k = 0 -> just t_n, the ego term)
// ---------------------------------------------------------------------------
__global__ __launch_bounds__(256) void scale_out_kernel(
    const float* __restrict__ user_t, const float* __restrict__ item_t,
    float4* __restrict__ out, int n_users, long long total_f4) {
  long long g = (long long)blockIdx.x * blockDim.x + threadIdx.x;
  if (g >= total_f4) return;
  int node  = (int)(g >> 7);
  int slice = (int)((g >> 5) & 3);
  float t     = (node < n_users) ? user_t[node] : item_t[node - n_users];
  float decay = 1.0f - t;
  float f     = t;
#pragma unroll
  for (int k = 0; k < 3; ++k) f = (k < slice) ? f * decay : f;
  float4 v = out[g];
  v.x *= f; v.y *= f; v.z *= f; v.w *= f;
  out[g] = v;
}

// ---------------------------------------------------------------------------
extern "C" void kernel_launch(void* const* d_in, const int* in_sizes, int n_in,
                              void* d_out, int out_size, void* d_ws, size_t ws_size,
                              hipStream_t stream) {
  const float* user_e = (const float*)d_in[0];
  const float* item_e = (const float*)d_in[1];
  const float* user_t = (const float*)d_in[2];
  const float* item_t = (const float*)d_in[3];
  const float* vals   = (const float*)d_in[4];
  const int*   rows   = (const int*)d_in[5];
  const int*   cols   = (const int*)d_in[6];

  const int n_users = in_sizes[2];
  const int n_items = in_sizes[3];
  const int n_nodes = n_users + n_items;
  const int n_edges = in_sizes[4];
  float* out = (float*)d_out;

  const long long total_f4   = (long long)n_nodes * (NODE_STRIDE / 4);
  const int       blocks_elem = (int)((total_f4 + 255) / 256);

  init_out_kernel<<<blocks_elem, 256, 0, stream>>>(
      (const float4*)user_e, (const float4*)item_e, (float4*)out,
      n_users, total_f4);

  const int n_chunks = (n_edges + CHUNK - 1) / CHUNK;
  const int blocks_s = n_chunks < 3840 ? n_chunks : 3840;  // ~5 chunks/block -> pipelined
  for (int k = 1; k <= 3; ++k) {
    spmm_scatter_kernel<<<blocks_s, 256, 0, stream>>>(
        rows, cols, vals,
        out + (size_t)(k - 1) * DIM,   // gather slice k-1 (unscaled)
        out + (size_t)k * DIM,         // accumulate slice k
        n_edges, n_chunks);
  }

  scale_out_kernel<<<blocks_elem, 256, 0, stream>>>(
      user_t, item_t, (float4*)out, n_users, total_f4);
}